// Cross_attention_74019466379811
// MI455X (gfx1250) — compile-verified
//
#include <hip/hip_runtime.h>
#include <math.h>

typedef float v2f __attribute__((ext_vector_type(2)));
typedef float v8f __attribute__((ext_vector_type(8)));

#define B_      8
#define C_      192
#define N_      16384
#define HEADS_  6
#define TRI_    325      // upper triangle of 25x25 tile grid
#define CHUNKS_ 4
#define CHUNKK_ 4096     // 16384 / 4
#define LDS_    400      // leading dim of padded symmetric S (25*16)

// ---------------- workspace layout (float offsets) ----------------
constexpr size_t SZ_PART = (size_t)CHUNKS_ * B_ * TRI_ * 256; // 2,662,400
constexpr size_t SZ_S    = (size_t)B_ * LDS_ * LDS_;          // 1,280,000
constexpr size_t SZ_P    = (size_t)B_ * C_ * C_;              //   294,912
constexpr size_t SZ_V    = (size_t)B_ * C_;                   //     1,536
constexpr size_t SZ_ATT  = (size_t)B_ * HEADS_ * 32 * 32;     //    49,152

constexpr size_t OFF_PART = 0;
constexpr size_t OFF_S    = OFF_PART + SZ_PART;
constexpr size_t OFF_PQ   = OFF_S    + SZ_S;
constexpr size_t OFF_PK   = OFF_PQ   + SZ_P;
constexpr size_t OFF_PQK  = OFF_PK   + SZ_P;
constexpr size_t OFF_RQ   = OFF_PQK  + SZ_P;
constexpr size_t OFF_RK   = OFF_RQ   + SZ_V;
constexpr size_t OFF_VQ   = OFF_RK   + SZ_V;
constexpr size_t OFF_VK   = OFF_VQ   + SZ_V;
constexpr size_t OFF_ATT  = OFF_VK   + SZ_V;
constexpr size_t OFF_WM   = OFF_ATT  + SZ_ATT;
constexpr size_t OFF_WEFF = OFF_WM   + SZ_P;
constexpr size_t OFF_BE   = OFF_WEFF + SZ_P;

__device__ __forceinline__ v8f wmma4(v2f a, v2f b, v8f c) {
  // D = A(16x4 f32) * B(4x16 f32) + C ; emits v_wmma_f32_16x16x4_f32
  return __builtin_amdgcn_wmma_f32_16x16x4_f32(false, a, false, b, (short)0, c,
                                               false, false);
}

// ================= K1: Gram of Z = [q; k; ones] per batch =================
// Z is 385x16384 (padded to 400). Computes upper-triangle 16x16 tiles of ZZ^T,
// K split into CHUNKS_ deterministic chunks written to a partial buffer.
// Rows 0..191 = q, 192..383 = k, 384 = ones, 385..399 = zeros.
// Divergence-free: tiles with j < 24 load unconditionally; only the 25 edge
// tiles (j == 24) use a branchless per-lane select after a safe dummy load.
__global__ __launch_bounds__(256) void gram_kernel(const float* __restrict__ q,
                                                   const float* __restrict__ k,
                                                   float* __restrict__ part) {
  int wave = blockIdx.x * 8 + (threadIdx.x >> 5);   // 10400 waves exactly
  int lane = threadIdx.x & 31;
  int chunk = wave & (CHUNKS_ - 1);
  int rest = wave >> 2;
  int t = rest % TRI_;
  int b = rest / TRI_;
  int i = 0, tt = t;                                 // tile (i,j), j>=i
  while (tt >= (25 - i)) { tt -= 25 - i; ++i; }
  int j = i + tt;

  int lr  = lane & 15;
  int off = (lane >> 4) << 1;                        // K offset 0 or 2
  int rowA = i * 16 + lr;
  int rowB = j * 16 + lr;

  const float* baseq = q + (size_t)b * C_ * N_;
  const float* basek = k + (size_t)b * C_ * N_;

  // per-lane row pointers (dummy = q row 0 for non-load lanes; safe memory)
  bool aLoad = rowA < 2 * C_;
  bool bLoad = rowB < 2 * C_;
  float cA = (rowA == 2 * C_) ? 1.f : 0.f;
  float cB = (rowB == 2 * C_) ? 1.f : 0.f;
  const float* pa = aLoad ? ((rowA < C_) ? baseq + (size_t)rowA * N_
                                         : basek + (size_t)(rowA - C_) * N_)
                          : baseq;
  const float* pb = bLoad ? ((rowB < C_) ? baseq + (size_t)rowB * N_
                                         : basek + (size_t)(rowB - C_) * N_)
                          : baseq;
  pa += off;
  pb += off;

  v8f acc0 = {0.f, 0.f, 0.f, 0.f, 0.f, 0.f, 0.f, 0.f};
  v8f acc1 = {0.f, 0.f, 0.f, 0.f, 0.f, 0.f, 0.f, 0.f};
  int kb = chunk * CHUNKK_;

  if (j < 24) {                                      // wave-uniform branch
    // -------- clean path: all lanes load, no selects --------
#pragma unroll 8
    for (int kk = kb; kk < kb + CHUNKK_; kk += 4) {
      v2f a  = *(const v2f*)(pa + kk);
      v2f bb = *(const v2f*)(pb + kk);
      if ((kk & 4) == 0) acc0 = wmma4(a, bb, acc0);
      else               acc1 = wmma4(a, bb, acc1);
    }
  } else {
    // -------- edge path: unconditional dummy load + cndmask select --------
#pragma unroll 8
    for (int kk = kb; kk < kb + CHUNKK_; kk += 4) {
      v2f la = *(const v2f*)(pa + kk);
      v2f lb = *(const v2f*)(pb + kk);
      v2f a, bb;
      a[0]  = aLoad ? la[0] : cA;
      a[1]  = aLoad ? la[1] : cA;
      bb[0] = bLoad ? lb[0] : cB;
      bb[1] = bLoad ? lb[1] : cB;
      if ((kk & 4) == 0) acc0 = wmma4(a, bb, acc0);
      else               acc1 = wmma4(a, bb, acc1);
    }
  }
  v8f acc;
#pragma unroll
  for (int r = 0; r < 8; ++r) acc[r] = acc0[r] + acc1[r];

  float* o = part + ((((size_t)chunk * B_ + b) * TRI_ + t) << 8);
  int mbase = (lane >> 4) << 3;                      // VGPR r -> M = r (+8 hi)
#pragma unroll
  for (int r = 0; r < 8; ++r) o[(mbase + r) * 16 + lr] = acc[r];
}

// ============ K2: reduce chunk partials, scatter into padded symmetric S ====
__global__ __launch_bounds__(256) void reduce_kernel(const float* __restrict__ part,
                                                     float* __restrict__ Sfull) {
  int e = blockIdx.x * 256 + threadIdx.x;            // 8*325*256 exactly
  int b = e / (TRI_ * 256);
  int r = e - b * TRI_ * 256;
  int t = r >> 8;
  int el = r & 255;
  int i = 0, tt = t;
  while (tt >= (25 - i)) { tt -= 25 - i; ++i; }
  int j = i + tt;
  float s = 0.f;
#pragma unroll
  for (int ch = 0; ch < CHUNKS_; ++ch)
    s += part[((((size_t)ch * B_ + b) * TRI_ + t) << 8) + el];
  int ml = el >> 4, nl = el & 15;
  int row = i * 16 + ml, col = j * 16 + nl;
  float* S = Sfull + (size_t)b * LDS_ * LDS_;
  S[(size_t)row * LDS_ + col] = s;
  if (i != j) S[(size_t)col * LDS_ + row] = s;
}

// ============ K3: generic 192x192x192 GEMM, per-batch strides ==============
__global__ __launch_bounds__(256) void gemm192(const float* __restrict__ A, long sA,
                                               const float* __restrict__ Bm, int ldb,
                                               long sB, float* __restrict__ Cm,
                                               long sC) {
  const float* Ab = A  + (size_t)blockIdx.y * sA;   // lda fixed = 192
  const float* Bb = Bm + (size_t)blockIdx.y * sB;
  float* Cb       = Cm + (size_t)blockIdx.y * sC;   // ldc fixed = 192
  int ti = blockIdx.x / 6, tj = blockIdx.x % 6;
  __shared__ float As[32][33], Bs[32][33];
  int tx = threadIdx.x & 31, ty = threadIdx.x >> 5;
  float acc[4] = {0.f, 0.f, 0.f, 0.f};
  for (int k0 = 0; k0 < C_; k0 += 32) {
#pragma unroll
    for (int r = 0; r < 4; ++r) {
      As[ty + 8 * r][tx] = Ab[(size_t)(ti * 32 + ty + 8 * r) * C_ + k0 + tx];
      Bs[ty + 8 * r][tx] = Bb[(size_t)(k0 + ty + 8 * r) * ldb + tj * 32 + tx];
    }
    __syncthreads();
#pragma unroll
    for (int kk = 0; kk < 32; ++kk) {
      float bvv = Bs[kk][tx];
#pragma unroll
      for (int r = 0; r < 4; ++r) acc[r] += As[ty + 8 * r][kk] * bvv;
    }
    __syncthreads();
  }
#pragma unroll
  for (int r = 0; r < 4; ++r)
    Cb[(size_t)(ti * 32 + ty + 8 * r) * C_ + tj * 32 + tx] = acc[r];
}

// ============ K4: per-channel inverse norms + bias-projection terms ========
__global__ void norms_kernel(const float* __restrict__ Sfull,
                             const float* __restrict__ Pq, const float* __restrict__ Pk,
                             const float* __restrict__ wq, const float* __restrict__ wk,
                             const float* __restrict__ bq, const float* __restrict__ bk,
                             float* __restrict__ rq, float* __restrict__ rk,
                             float* __restrict__ vq, float* __restrict__ vk) {
  int b = blockIdx.x / C_;
  int c = blockIdx.x % C_;
  int lane = threadIdx.x;                            // block = 32
  const float* S  = Sfull + (size_t)b * LDS_ * LDS_;
  const float* pq = Pq + (size_t)b * C_ * C_ + (size_t)c * C_;
  const float* pk = Pk + (size_t)b * C_ * C_ + (size_t)c * C_;
  float dq = 0.f, aq = 0.f, dk = 0.f, ak = 0.f;
  for (int j2 = lane; j2 < C_; j2 += 32) {
    float wqv = wq[c * C_ + j2], wkv = wk[c * C_ + j2];
    dq += pq[j2] * wqv;
    aq += wqv * S[(size_t)j2 * LDS_ + 2 * C_];             // (Wq @ rowsum(q))[c]
    dk += pk[j2] * wkv;
    ak += wkv * S[(size_t)(C_ + j2) * LDS_ + 2 * C_];      // (Wk @ rowsum(k))[c]
  }
  for (int m = 16; m >= 1; m >>= 1) {
    dq += __shfl_xor(dq, m); aq += __shfl_xor(aq, m);
    dk += __shfl_xor(dk, m); ak += __shfl_xor(ak, m);
  }
  if (lane == 0) {
    float bqc = bq[c], bkc = bk[c];
    float nq = dq + 2.f * bqc * aq + (float)N_ * bqc * bqc;
    float nk = dk + 2.f * bkc * ak + (float)N_ * bkc * bkc;
    rq[b * C_ + c] = 1.f / fmaxf(sqrtf(fmaxf(nq, 0.f)), 1e-12f);
    rk[b * C_ + c] = 1.f / fmaxf(sqrtf(fmaxf(nk, 0.f)), 1e-12f);
    vq[b * C_ + c] = aq;
    vk[b * C_ + c] = ak;
  }
}

// ============ K5: per-head logits + softmax (one wave per attn row) ========
__global__ __launch_bounds__(256) void softmax_kernel(
    const float* __restrict__ Pqk, const float* __restrict__ wk,
    const float* __restrict__ bq, const float* __restrict__ bk,
    const float* __restrict__ vq, const float* __restrict__ vk,
    const float* __restrict__ rq, const float* __restrict__ rk,
    const float* __restrict__ temp, float* __restrict__ attn) {
  int wv = blockIdx.x * 8 + (threadIdx.x >> 5);      // 1536 rows exactly
  int lane = threadIdx.x & 31;
  int c = wv & 31;
  int h = (wv >> 5) % HEADS_;
  int b = wv / (32 * HEADS_);
  int cp = h * 32 + c, dp = h * 32 + lane;
  const float* prow = Pqk + (size_t)b * C_ * C_ + (size_t)cp * C_;
  const float* wrow = wk + (size_t)dp * C_;
  float dot = 0.f;
  for (int j2 = 0; j2 < C_; ++j2) dot += prow[j2] * wrow[j2];
  float logit = dot + vq[b * C_ + cp] * bk[dp] + bq[cp] * vk[b * C_ + dp] +
                (float)N_ * bq[cp] * bk[dp];
  float val = logit * rq[b * C_ + cp] * rk[b * C_ + dp] * temp[h];
  float m = val;
  for (int s = 16; s >= 1; s >>= 1) m = fmaxf(m, __shfl_xor(m, s));
  float e = __expf(val - m);
  float sum = e;
  for (int s = 16; s >= 1; s >>= 1) sum += __shfl_xor(sum, s);
  attn[(size_t)wv * 32 + lane] = e / sum;
}

// ============ K6: Wm = Wo @ blockdiag(attn) ================================
__global__ __launch_bounds__(256) void wm_kernel(const float* __restrict__ attn,
                                                 const float* __restrict__ wo,
                                                 float* __restrict__ Wm) {
  int idx = blockIdx.x * 256 + threadIdx.x;          // 8*192*192 exactly
  int b = idx / (C_ * C_);
  int rem = idx - b * C_ * C_;
  int o = rem / C_;
  int jc = rem - o * C_;
  int h = jc >> 5, d = jc & 31;
  const float* at = attn + (((size_t)b * HEADS_ + h) << 10) + d;
  const float* wr = wo + (size_t)o * C_ + h * 32;
  float s = 0.f;
#pragma unroll
  for (int c2 = 0; c2 < 32; ++c2) s += wr[c2] * at[(size_t)c2 * 32];
  Wm[idx] = s;
}

// ============ K7: beff = Wm @ bv + bo ======================================
__global__ void beff_kernel(const float* __restrict__ Wm, const float* __restrict__ bv,
                            const float* __restrict__ bo, float* __restrict__ beff) {
  int idx = blockIdx.x * 256 + threadIdx.x;          // 1536 exactly
  int b = idx / C_, o = idx % C_;
  const float* wr = Wm + (size_t)b * C_ * C_ + (size_t)o * C_;
  float s = bo[o];
  for (int d = 0; d < C_; ++d) s += wr[d] * bv[d];
  beff[idx] = s;
}

// ============ K8: out = Weff @ v + beff (WMMA, 16x64 strip per wave) =======
__global__ __launch_bounds__(256) void out_gemm(const float* __restrict__ vin,
                                                const float* __restrict__ Weff,
                                                const float* __restrict__ beff,
                                                float* __restrict__ out) {
  int wave = blockIdx.x * 8 + (threadIdx.x >> 5);    // 8*12*256 waves exactly
  int lane = threadIdx.x & 31;
  int ng = wave & 255;
  int mt = (wave >> 8) % 12;
  int b  = (wave >> 8) / 12;
  int lr = lane & 15, hi = lane >> 4, off = hi << 1;
  const float* A  = Weff + (size_t)b * C_ * C_ + (size_t)(mt * 16 + lr) * C_ + off;
  const float* Bv = vin + (size_t)b * C_ * N_ + (size_t)off * N_ + ng * 64 + lr;
  v8f acc[4];
#pragma unroll
  for (int nt = 0; nt < 4; ++nt)
#pragma unroll
    for (int r = 0; r < 8; ++r) acc[nt][r] = 0.f;
#pragma unroll 4
  for (int kk = 0; kk < C_; kk += 4) {
    v2f a = *(const v2f*)(A + kk);
    const float* bp = Bv + (size_t)kk * N_;
#pragma unroll
    for (int nt = 0; nt < 4; ++nt) {
      v2f f;
      f[0] = bp[nt * 16];
      f[1] = bp[N_ + nt * 16];
      acc[nt] = wmma4(a, f, acc[nt]);
    }
  }
  int mb = mt * 16 + hi * 8;
  float be[8];
#pragma unroll
  for (int r = 0; r < 8; ++r) be[r] = beff[b * C_ + mb + r];
  float* o = out + ((size_t)b * C_ + mb) * N_ + ng * 64 + lr;
#pragma unroll
  for (int nt = 0; nt < 4; ++nt)
#pragma unroll
    for (int r = 0; r < 8; ++r)
      o[(size_t)r * N_ + nt * 16] = acc[nt][r] + be[r];
}

// =========================== host launcher =================================
extern "C" void kernel_launch(void* const* d_in, const int* in_sizes, int n_in,
                              void* d_out, int out_size, void* d_ws, size_t ws_size,
                              hipStream_t stream) {
  (void)in_sizes; (void)n_in; (void)out_size; (void)ws_size;
  const float* q    = (const float*)d_in[0];
  const float* k    = (const float*)d_in[1];
  const float* v    = (const float*)d_in[2];
  const float* wq   = (const float*)d_in[3];
  const float* bq   = (const float*)d_in[4];
  const float* wk   = (const float*)d_in[5];
  const float* bk   = (const float*)d_in[6];
  const float* wv   = (const float*)d_in[7];
  const float* bv   = (const float*)d_in[8];
  const float* wo   = (const float*)d_in[9];
  const float* bo   = (const float*)d_in[10];
  const float* temp = (const float*)d_in[11];
  float* out = (float*)d_out;
  float* w   = (float*)d_ws;

  float* part  = w + OFF_PART;
  float* Sfull = w + OFF_S;
  float* Pq    = w + OFF_PQ;
  float* Pk    = w + OFF_PK;
  float* Pqk   = w + OFF_PQK;
  float* rq    = w + OFF_RQ;
  float* rk    = w + OFF_RK;
  float* vq    = w + OFF_VQ;
  float* vk    = w + OFF_VK;
  float* attn  = w + OFF_ATT;
  float* Wm    = w + OFF_WM;
  float* Weff  = w + OFF_WEFF;
  float* beff  = w + OFF_BE;

  gram_kernel<<<1300, 256, 0, stream>>>(q, k, part);
  reduce_kernel<<<2600, 256, 0, stream>>>(part, Sfull);
  // Pq = Wq @ Sqq ; Pk = Wk @ Skk ; Pqk = Wq @ Sqk
  gemm192<<<dim3(36, B_), 256, 0, stream>>>(wq, 0, Sfull, LDS_,
                                            (long)LDS_ * LDS_, Pq, (long)C_ * C_);
  gemm192<<<dim3(36, B_), 256, 0, stream>>>(wk, 0, Sfull + (size_t)C_ * LDS_ + C_,
                                            LDS_, (long)LDS_ * LDS_, Pk,
                                            (long)C_ * C_);
  gemm192<<<dim3(36, B_), 256, 0, stream>>>(wq, 0, Sfull + C_, LDS_,
                                            (long)LDS_ * LDS_, Pqk, (long)C_ * C_);
  norms_kernel<<<B_ * C_, 32, 0, stream>>>(Sfull, Pq, Pk, wq, wk, bq, bk,
                                           rq, rk, vq, vk);
  softmax_kernel<<<192, 256, 0, stream>>>(Pqk, wk, bq, bk, vq, vk, rq, rk,
                                          temp, attn);
  wm_kernel<<<1152, 256, 0, stream>>>(attn, wo, Wm);
  // Weff = Wm @ Wv
  gemm192<<<dim3(36, B_), 256, 0, stream>>>(Wm, (long)C_ * C_, wv, C_, 0,
                                            Weff, (long)C_ * C_);
  beff_kernel<<<6, 256, 0, stream>>>(Wm, bv, bo, beff);
  out_gemm<<<3072, 256, 0, stream>>>(v, Weff, beff, out);
}